// TFLOPLoss_50568944943580
// MI455X (gfx1250) — compile-verified
//
#include <hip/hip_runtime.h>
#include <hip/hip_bf16.h>
#include <stdint.h>

// Problem constants from the reference: B, L, M, C = (32, 688, 4, 32)
#define BB 32
#define LL 688
#define MM 4
#define CC 32
#define ROWS_PER_BLOCK 16
#define TPAD 692                         // padded LDS row stride (floats); 692%64=52 -> conflict-free column access
#define TILES_PER_B (LL / ROWS_PER_BLOCK)      // 43
#define NBLOCKS (BB * TILES_PER_B)             // 1376
#define CHUNKS_PER_ROW (LL * 4 / 16)           // 172 x 16B per row
#define CHUNKS (ROWS_PER_BLOCK * CHUNKS_PER_ROW)  // 2752

typedef __attribute__((ext_vector_type(2))) float v2f;
typedef __attribute__((ext_vector_type(8))) float v8f;
typedef __attribute__((ext_vector_type(4))) int   v4i;

// ---------------- CDNA5 async global->LDS (ASYNCcnt path) ----------------
__device__ __forceinline__ void async_g2l_b128(void* lds_generic, const void* gaddr) {
#if __has_builtin(__builtin_amdgcn_global_load_async_to_lds_b128)
  __builtin_amdgcn_global_load_async_to_lds_b128(
      (__attribute__((address_space(1))) v4i*)(void*)gaddr,
      (__attribute__((address_space(3))) v4i*)lds_generic,
      0, 0);
#else
  uint32_t l = (uint32_t)(uintptr_t)(__attribute__((address_space(3))) void*)lds_generic;
  asm volatile("global_load_async_to_lds_b128 %0, %1, off"
               :: "v"(l), "v"(gaddr) : "memory");
#endif
}

__device__ __forceinline__ void wait_async0() {
#if __has_builtin(__builtin_amdgcn_s_wait_asynccnt)
  __builtin_amdgcn_s_wait_asynccnt(0);
#else
  asm volatile("s_wait_asynccnt 0" ::: "memory");
#endif
}

// ---------------- wave32 reductions ----------------
__device__ __forceinline__ float wave_max32(float v) {
#pragma unroll
  for (int off = 16; off >= 1; off >>= 1) v = fmaxf(v, __shfl_xor(v, off, 32));
  return v;
}
__device__ __forceinline__ float wave_sum32(float v) {
#pragma unroll
  for (int off = 16; off >= 1; off >>= 1) v += __shfl_xor(v, off, 32);
  return v;
}

// =====================================================================
// Kernel 1: one block per 16-row tile of pointer_logits (same batch b).
// Produces 5 per-block partials: {ptr_sum, ptr_cnt, cls_sum, bce_sum, vm_sum}
// =====================================================================
__global__ __launch_bounds__(256)
void tflop_loss_main(const float* __restrict__ tag_logits,
                     const float* __restrict__ pointer_logits,
                     const float* __restrict__ empty_logits,
                     const int* __restrict__ tag_targets,
                     const int* __restrict__ box_indices,
                     const unsigned char* __restrict__ data_tag_mask,
                     const unsigned char* __restrict__ attention_mask,
                     float* __restrict__ partials)
{
  __shared__ float tile[ROWS_PER_BLOCK * TPAD];    // 44288 B
  __shared__ float smaskf[LL];                     // seq mask as float (branchless multiply)
  __shared__ float rowmax[ROWS_PER_BLOCK];
  __shared__ float rowsum[ROWS_PER_BLOCK];
  __shared__ float accs[5];

  const int tid   = threadIdx.x;
  const int lane  = tid & 31;
  const int wave  = tid >> 5;
  const int blk   = blockIdx.x;
  const int b     = blk / TILES_PER_B;
  const int ibase = (blk % TILES_PER_B) * ROWS_PER_BLOCK;

  if (tid < 5) accs[tid] = 0.0f;
  if (tid < ROWS_PER_BLOCK) rowsum[tid] = 0.0f;

  // seq_mask = data_tag_mask[:, L:] as floats (0.0 / 1.0)
  for (int j = tid; j < LL; j += 256)
    smaskf[j] = (float)data_tag_mask[b * (2 * LL) + LL + j];

  // ---- async copy 16x688 f32 tile into padded LDS (16B chunks, coalesced) ----
  const char* gbase = (const char*)(pointer_logits + (size_t)(b * LL + ibase) * LL);
  for (int c = tid; c < CHUNKS; c += 256) {
    const int row = c / CHUNKS_PER_ROW;
    const int off = (c % CHUNKS_PER_ROW) * 16;     // bytes within row
    async_g2l_b128((char*)&tile[row * TPAD] + off, gbase + (size_t)row * (LL * 4) + off);
  }
  wait_async0();
  __syncthreads();

  // ---- Phase B: per-row max (wave w handles rows 2w, 2w+1) ----
#pragma unroll
  for (int rr = 0; rr < 2; ++rr) {
    const int r = wave * 2 + rr;
    float m = -3.4e38f;
#pragma unroll
    for (int t = 0; t < 22; ++t) {
      const int idx = lane + 32 * t;
      if (idx < LL) m = fmaxf(m, tile[r * TPAD + idx]);
    }
    m = wave_max32(m);
    if (lane == 0) rowmax[r] = m;
  }
  __syncthreads();

  // ---- Phase C: masked exp row-sums via V_WMMA_F32_16X16X4_F32 ----
  // B[k][n] = exp(tile[n][k]-mx[n]) * seq_mask[k]; A = ones(16x4) -> D accumulates row sums.
  // Branchless: mask applied by multiply; tile & mask read as ds_load_b64 (8B aligned).
  {
    const int kbeg = wave * 88;                    // 88 % 8 == 0; wave7 gets 616..688 (72, also %8==0)
    const int kend = (kbeg + 88 < LL) ? (kbeg + 88) : LL;
    const int n  = lane & 15;
    const int hi = lane >> 4;
    const float mx = rowmax[n];
    const float* trow = &tile[n * TPAD];
#if __has_builtin(__builtin_amdgcn_wmma_f32_16x16x4_f32)
    v8f acc = {0.f, 0.f, 0.f, 0.f, 0.f, 0.f, 0.f, 0.f};
    const v2f aOnes = {1.0f, 1.0f};
    for (int k0 = kbeg; k0 < kend; k0 += 8) {
      const int ka = k0 + hi * 2;
      const v2f x0 = *(const v2f*)&trow[ka];
      const v2f m0 = *(const v2f*)&smaskf[ka];
      const v2f x1 = *(const v2f*)&trow[ka + 4];
      const v2f m1 = *(const v2f*)&smaskf[ka + 4];
      v2f b0, b1;
      b0[0] = __expf(x0[0] - mx) * m0[0];
      b0[1] = __expf(x0[1] - mx) * m0[1];
      b1[0] = __expf(x1[0] - mx) * m1[0];
      b1[1] = __expf(x1[1] - mx) * m1[1];
      acc = __builtin_amdgcn_wmma_f32_16x16x4_f32(false, aOnes, false, b0,
                                                  (short)0, acc, false, false);
      acc = __builtin_amdgcn_wmma_f32_16x16x4_f32(false, aOnes, false, b1,
                                                  (short)0, acc, false, false);
    }
    const float part = acc[0];                     // D[M=0][N=n] on lanes 0..15
    if (lane < 16) atomicAdd(&rowsum[n], part);
#else
    float s = 0.0f;
    for (int k0 = kbeg; k0 < kend; k0 += 8) {
      const int ka = k0 + hi * 2;
      const v2f x0 = *(const v2f*)&trow[ka];
      const v2f m0 = *(const v2f*)&smaskf[ka];
      const v2f x1 = *(const v2f*)&trow[ka + 4];
      const v2f m1 = *(const v2f*)&smaskf[ka + 4];
      s += __expf(x0[0] - mx) * m0[0] + __expf(x0[1] - mx) * m0[1];
      s += __expf(x1[0] - mx) * m1[0] + __expf(x1[1] - mx) * m1[1];
    }
    s += __shfl_xor(s, 16, 32);                    // fold hi-half K slots into lo lanes
    if (lane < 16) atomicAdd(&rowsum[n], s);
#endif
  }
  __syncthreads();

  // ---- Phase D: gather <=4 deduped targets per row, accumulate ptr loss ----
  if (tid < ROWS_PER_BLOCK * MM) {
    const int r = tid >> 2;
    const int m = tid & 3;
    const int i = ibase + r;
    const int* bx = box_indices + (size_t)(b * LL + i) * MM;
    const int raw = bx[m];
    if (raw != -1) {
      int tgt = raw < 0 ? 0 : (raw > LL - 1 ? LL - 1 : raw);
      bool first = true;
      for (int mm = 0; mm < m; ++mm) {
        const int o = bx[mm];
        if (o != -1) {
          int ot = o < 0 ? 0 : (o > LL - 1 ? LL - 1 : o);
          if (ot == tgt) { first = false; break; }
        }
      }
      if (first && smaskf[tgt] != 0.0f) {
        const float denom = rowsum[r] + 1e-10f;
        const float p = __expf(tile[r * TPAD + tgt] - rowmax[r]) / denom;
        const float loss = -__logf(p + 1e-10f);
        atomicAdd(&accs[0], loss);
        if (loss != 0.0f) atomicAdd(&accs[1], 1.0f);
      }
    }
  }

  // ---- Phase E1: tag classification loss (C=32 -> one lane per class) ----
#pragma unroll
  for (int rr = 0; rr < 2; ++rr) {
    const int g = b * LL + ibase + wave * 2 + rr;
    const float x = tag_logits[(size_t)g * CC + lane];
    const float mx = wave_max32(x);
    const float s = wave_sum32(__expf(x - mx));
    const int t = tag_targets[g];
    if (lane == t) atomicAdd(&accs[2], -(x - mx - __logf(s)));
  }

  // ---- Phase E2: empty-pointer BCE for this tile's 16 positions ----
  if (tid < ROWS_PER_BLOCK) {
    const int i = ibase + tid;
    const float x = empty_logits[b * LL + i];
    const unsigned char a_ = attention_mask[b * (2 * LL) + LL + i];
    const unsigned char d_ = data_tag_mask[b * (2 * LL) + LL + i];
    const float tf = (a_ && !d_) ? 1.0f : 0.0f;
    const float bce = fmaxf(x, 0.0f) - x * tf + log1pf(__expf(-fabsf(x)));
    const float vm = a_ ? 1.0f : 0.0f;
    atomicAdd(&accs[3], bce * vm);
    atomicAdd(&accs[4], vm);
  }
  __syncthreads();

  if (tid < 5) partials[(size_t)blk * 5 + tid] = accs[tid];
}

// =====================================================================
// Kernel 2: deterministic reduction of 1376x5 partials -> 4 outputs
// =====================================================================
__global__ __launch_bounds__(256)
void tflop_loss_finalize(const float* __restrict__ partials, float* __restrict__ out)
{
  __shared__ float red[5][256];
  const int tid = threadIdx.x;
  float a[5] = {0.f, 0.f, 0.f, 0.f, 0.f};
  for (int i = tid; i < NBLOCKS; i += 256) {
#pragma unroll
    for (int j = 0; j < 5; ++j) a[j] += partials[(size_t)i * 5 + j];
  }
#pragma unroll
  for (int j = 0; j < 5; ++j) red[j][tid] = a[j];
  __syncthreads();
  for (int s = 128; s > 0; s >>= 1) {
    if (tid < s) {
#pragma unroll
      for (int j = 0; j < 5; ++j) red[j][tid] += red[j][tid + s];
    }
    __syncthreads();
  }
  if (tid == 0) {
    const float ptr_sum = red[0][0];
    const float cnt     = red[1][0];
    const float cls     = red[2][0] / (float)(BB * LL);
    const float ptr     = (cnt == 0.0f) ? 0.0f : ptr_sum / fmaxf(cnt, 1.0f);
    const float emp     = red[3][0] / fmaxf(red[4][0], 1.0f);
    out[0] = cls + ptr + emp;
    out[1] = cls;
    out[2] = ptr;
    out[3] = emp;
  }
}

extern "C" void kernel_launch(void* const* d_in, const int* in_sizes, int n_in,
                              void* d_out, int out_size, void* d_ws, size_t ws_size,
                              hipStream_t stream) {
  const float* tag_logits  = (const float*)d_in[0];
  const float* ptr_logits  = (const float*)d_in[1];
  const float* empty_log   = (const float*)d_in[2];
  const int*   tag_targets = (const int*)d_in[3];
  const int*   box_indices = (const int*)d_in[4];
  const unsigned char* dtm = (const unsigned char*)d_in[5];   // bool arrays: 1 byte/elem
  const unsigned char* att = (const unsigned char*)d_in[6];
  float* partials = (float*)d_ws;                              // NBLOCKS*5 floats = 27.5 KB
  float* out      = (float*)d_out;                             // 4 floats

  tflop_loss_main<<<NBLOCKS, 256, 0, stream>>>(tag_logits, ptr_logits, empty_log,
                                               tag_targets, box_indices, dtm, att,
                                               partials);
  tflop_loss_finalize<<<1, 256, 0, stream>>>(partials, out);
}